// GLMBP_4045859193635
// MI455X (gfx1250) — compile-verified
//
#include <hip/hip_runtime.h>
#include <math.h>

#define BSZ  8
#define CIN  256
#define COUTC 256
#define HH   64
#define WWI  64
#define LL   4096
#define DMc  64
#define DIc  128
#define DSc  16
#define DCc  4
#define DTRc 4
#define NROWS (BSZ * LL)          // 32768 token rows
#define NFUSE (DIc + 2 * DSc)     // 160
#define SSEG 32                   // scan segments
#define LSEG (LL / SSEG)          // 128 steps per segment
#define EPSF 1e-5f

typedef float v2f __attribute__((ext_vector_type(2)));
typedef float v8f __attribute__((ext_vector_type(8)));

__device__ __forceinline__ float siluf(float x) { return x / (1.f + __expf(-x)); }
__device__ __forceinline__ float softplusf(float x) { return (x > 20.f) ? x : log1pf(__expf(x)); }
__device__ __forceinline__ float geluf(float x) { return 0.5f * x * (1.f + erff(x * 0.70710678118654752f)); }

// ---------------------------------------------------------------------------
// LayerNorm over C=256 for each (b,l). Strides parameterized: handles both
// x [B,C,L] (cStride=L) and xm [B,L,C] (cStride=1). Output [B,L,C].
// ---------------------------------------------------------------------------
__global__ void k_ln(const float* __restrict__ in, const float* __restrict__ g,
                     const float* __restrict__ bvec, float* __restrict__ out,
                     int bStride, int cStride, int lStride) {
  int t = blockIdx.x * blockDim.x + threadIdx.x;
  if (t >= NROWS) return;
  int b = t >> 12;            // / LL
  int l = t & (LL - 1);
  const float* p = in + (size_t)b * bStride + (size_t)l * lStride;
  float s = 0.f, s2 = 0.f;
  for (int c = 0; c < CIN; ++c) {
    float v = p[(size_t)c * cStride];
    s += v; s2 += v * v;
  }
  float m  = s * (1.f / CIN);
  float va = fmaxf(s2 * (1.f / CIN) - m * m, 0.f);
  float rs = rsqrtf(va + EPSF);
  float* o = out + (size_t)t * CIN;
  for (int c = 0; c < CIN; ++c) {
    float v = p[(size_t)c * cStride];
    o[c] = (v - m) * rs * g[c] + bvec[c];
  }
}

// ---------------------------------------------------------------------------
// Weight prep: flipped in/out projections (folds the channel-flip branches of
// the bidirectional mamba into weights) + fused dt/x projection matrix.
// ---------------------------------------------------------------------------
__global__ void k_prep(const float* __restrict__ in_proj_w, const float* __restrict__ out_proj_w,
                       const float* __restrict__ x_proj_w, const float* __restrict__ dt_proj_w,
                       const float* __restrict__ dt_proj_b,
                       float* __restrict__ WinF, float* __restrict__ WoF,
                       float* __restrict__ Wcomb, float* __restrict__ bcomb) {
  int t = blockIdx.x * blockDim.x + threadIdx.x;
  if (t < 2 * DIc * DMc) {
    int n = t / DMc, j = t % DMc;
    WinF[t] = in_proj_w[n * DMc + (DMc - 1 - j)];
    return;
  }
  t -= 2 * DIc * DMc;
  if (t < DMc * DIc) {
    int d = t / DIc, k = t % DIc;
    WoF[t] = out_proj_w[(DMc - 1 - d) * DIc + k];
    return;
  }
  t -= DMc * DIc;
  if (t < NFUSE * DIc) {
    int i = t / DIc, k = t % DIc;
    float v;
    if (i < DIc) {
      v = 0.f;
      for (int r = 0; r < DTRc; ++r) v += dt_proj_w[i * DTRc + r] * x_proj_w[r * DIc + k];
    } else if (i < DIc + DSc) {
      v = x_proj_w[(DTRc + (i - DIc)) * DIc + k];
    } else {
      v = x_proj_w[(DTRc + DSc + (i - DIc - DSc)) * DIc + k];
    }
    Wcomb[t] = v;
    return;
  }
  t -= NFUSE * DIc;
  if (t < NFUSE) bcomb[t] = (t < DIc) ? dt_proj_b[t] : 0.f;
}

// ---------------------------------------------------------------------------
// WMMA fp32 GEMM with register blocking: each wave computes an
// (MB*16) x (NB*16) macro-tile; per k-step MB+NB b64 loads feed MB*NB
// V_WMMA_F32_16X16X4_F32 ops. Guard is wave-uniform so EXEC is all-ones.
// flags: bit0 = +bias[n]; bit1 = accumulate into Out; bit2 = transposed store
//        Out[b, n, l] with m = b*LL + l (final projection only).
// ---------------------------------------------------------------------------
template <int MB, int NB>
__global__ void k_gemm_wmma(const float* __restrict__ A, const float* __restrict__ W,
                            const float* __restrict__ bias, float* __restrict__ Out,
                            int M, int N, int K, int lda, int ldw, int ldo, int flags) {
  int wid = blockIdx.x * (blockDim.x >> 5) + (threadIdx.x >> 5);
  int groups_n = (N >> 4) / NB;
  int gm = wid / groups_n;
  int gn = wid - gm * groups_n;
  int tm0 = gm * 16 * MB;
  int tn0 = gn * 16 * NB;
  if (tm0 >= M) return;                    // wave-uniform
  int lane   = threadIdx.x & 31;
  int lane16 = lane & 15;
  int half   = lane >> 4;

  const float* Arow[MB];
  const float* Wrow[NB];
#pragma unroll
  for (int i = 0; i < MB; ++i)
    Arow[i] = A + (size_t)(tm0 + i * 16 + lane16) * lda + 2 * half;
#pragma unroll
  for (int j = 0; j < NB; ++j)
    Wrow[j] = W + (size_t)(tn0 + j * 16 + lane16) * ldw + 2 * half;

  v8f acc[MB][NB];
#pragma unroll
  for (int i = 0; i < MB; ++i)
#pragma unroll
    for (int j = 0; j < NB; ++j)
      acc[i][j] = (v8f){0.f, 0.f, 0.f, 0.f, 0.f, 0.f, 0.f, 0.f};

  for (int k0 = 0; k0 < K; k0 += 4) {
    v2f a[MB], b[NB];
#pragma unroll
    for (int i = 0; i < MB; ++i) a[i] = *reinterpret_cast<const v2f*>(Arow[i] + k0);
#pragma unroll
    for (int j = 0; j < NB; ++j) b[j] = *reinterpret_cast<const v2f*>(Wrow[j] + k0);
#pragma unroll
    for (int i = 0; i < MB; ++i)
#pragma unroll
      for (int j = 0; j < NB; ++j)
        acc[i][j] = __builtin_amdgcn_wmma_f32_16x16x4_f32(false, a[i], false, b[j],
                                                          (short)0, acc[i][j],
                                                          false, false);
  }

#pragma unroll
  for (int j = 0; j < NB; ++j) {
    int n = tn0 + j * 16 + lane16;
    float bv = (flags & 1) ? bias[n] : 0.f;
#pragma unroll
    for (int i = 0; i < MB; ++i) {
#pragma unroll
      for (int r = 0; r < 8; ++r) {
        int m = tm0 + i * 16 + r + 8 * half;
        float v = acc[i][j][r] + bv;
        if (flags & 4) {
          int b_ = m >> 12;                // m / LL
          int l_ = m & (LL - 1);
          Out[((size_t)b_ * N + n) * LL + l_] = v;
        } else {
          size_t idx = (size_t)m * ldo + n;
          if (flags & 2) v += Out[idx];
          Out[idx] = v;
        }
      }
    }
  }
}

// ---------------------------------------------------------------------------
// Causal depthwise conv1d (DC=4) over L on the x-half of xz [B,L,256] + SiLU.
// ---------------------------------------------------------------------------
__global__ void k_conv1d(const float* __restrict__ xz, const float* __restrict__ w,
                         const float* __restrict__ cb, float* __restrict__ out) {
  int t = blockIdx.x * blockDim.x + threadIdx.x;
  if (t >= NROWS * DIc) return;
  int di = t & (DIc - 1);
  int bl = t >> 7;
  int l  = bl & (LL - 1);
  const float* src = xz + (size_t)bl * (2 * DIc) + di;
  float acc = cb[di];
#pragma unroll
  for (int k = 0; k < DCc; ++k) {
    int l2 = l - (DCc - 1) + k;
    if (l2 >= 0) acc += w[di * DCc + k] * src[(ptrdiff_t)(l2 - l) * (2 * DIc)];
  }
  out[t] = siluf(acc);
}

// ---------------------------------------------------------------------------
// Chunked parallel selective scan (linear recurrence h <- dA*h + dBu).
// Lane mapping everywhere: lane = ch*16 + ds, di = 2*dp + ch.
// Pass 1: per segment, cumulative transition P = prod(dA) and local terminal
//         state q (assuming h0 = 0).
// Pass 2: 32-step sequential prefix across segments -> true segment-entry
//         states H0.
// Pass 3: re-run each segment from H0 producing gated outputs.
// ---------------------------------------------------------------------------
__global__ void k_scan_p1(const float* __restrict__ xconv, const float* __restrict__ fused,
                          const float* __restrict__ A_log,
                          float* __restrict__ Pbuf, float* __restrict__ qbuf) {
  int wave = blockIdx.x;              // ((b*64 + dp) * SSEG + s)
  int s  = wave & (SSEG - 1);
  int wd = wave >> 5;
  int dp = wd & 63;
  int b  = wd >> 6;
  int lane = threadIdx.x;
  int ds = lane & 15;
  int ch = lane >> 4;
  int di = dp * 2 + ch;
  float Aval = -__expf(A_log[di * DSc + ds]);
  const float* fb = fused + ((size_t)b * LL + (size_t)s * LSEG) * NFUSE;
  const float* xb = xconv + ((size_t)b * LL + (size_t)s * LSEG) * DIc;
  float P = 1.f, h = 0.f;
  for (int l = 0; l < LSEG; ++l) {
    const float* f = fb + (size_t)l * NFUSE;
    float delta = softplusf(f[di]);
    float dA = __expf(delta * Aval);
    float Bv = f[DIc + ds];
    float xv = xb[(size_t)l * DIc + di];
    P *= dA;
    h = dA * h + (delta * Bv) * xv;
  }
  size_t idx = (((size_t)b * DIc + di) * DSc + ds) * SSEG + s;
  Pbuf[idx] = P;
  qbuf[idx] = h;
}

__global__ void k_scan_p2(const float* __restrict__ Pbuf, const float* __restrict__ qbuf,
                          float* __restrict__ H0) {
  int wave = blockIdx.x;              // b*64 + dp
  int dp = wave & 63;
  int b  = wave >> 6;
  int lane = threadIdx.x;
  int ds = lane & 15;
  int ch = lane >> 4;
  int di = dp * 2 + ch;
  size_t base = (((size_t)b * DIc + di) * DSc + ds) * SSEG;
  float h = 0.f;
  for (int s = 0; s < SSEG; ++s) {
    H0[base + s] = h;
    h = Pbuf[base + s] * h + qbuf[base + s];
  }
}

__global__ void k_scan_p3(const float* __restrict__ xconv, const float* __restrict__ fused,
                          const float* __restrict__ xz, const float* __restrict__ A_log,
                          const float* __restrict__ Dvec, const float* __restrict__ H0,
                          float* __restrict__ yg) {
  int wave = blockIdx.x;              // ((b*64 + dp) * SSEG + s)
  int s  = wave & (SSEG - 1);
  int wd = wave >> 5;
  int dp = wd & 63;
  int b  = wd >> 6;
  int lane = threadIdx.x;
  int ds = lane & 15;
  int ch = lane >> 4;
  int di = dp * 2 + ch;
  float Aval = -__expf(A_log[di * DSc + ds]);
  float Dv   = Dvec[di];
  size_t l0 = (size_t)s * LSEG;
  const float* fb = fused + ((size_t)b * LL + l0) * NFUSE;
  const float* xb = xconv + ((size_t)b * LL + l0) * DIc;
  const float* zb = xz    + ((size_t)b * LL + l0) * (2 * DIc) + DIc;
  float* yb = yg + ((size_t)b * LL + l0) * DIc;
  float h = H0[(((size_t)b * DIc + di) * DSc + ds) * SSEG + s];
  for (int l = 0; l < LSEG; ++l) {
    const float* f = fb + (size_t)l * NFUSE;
    float delta = softplusf(f[di]);
    float Bv = f[DIc + ds];
    float Cv = f[DIc + DSc + ds];
    float xv = xb[(size_t)l * DIc + di];
    h = __expf(delta * Aval) * h + (delta * Bv) * xv;
    float p = h * Cv;
    p += __shfl_xor(p, 1, 32);
    p += __shfl_xor(p, 2, 32);
    p += __shfl_xor(p, 4, 32);
    p += __shfl_xor(p, 8, 32);
    if (ds == 0) {
      float zv = zb[(size_t)l * (2 * DIc) + di];
      yb[(size_t)l * DIc + di] = (p + Dv * xv) * siluf(zv);
    }
  }
}

// ---------------------------------------------------------------------------
// Initialize skip terms: xm chunk0 = s*x1, chunk1 = s*x2, chunk3 = (1+s)*x4.
// ---------------------------------------------------------------------------
__global__ void k_init_xm(const float* __restrict__ xn, const float* __restrict__ skip,
                          float* __restrict__ xm) {
  int t = blockIdx.x * blockDim.x + threadIdx.x;   // NROWS * DM
  if (t >= NROWS * DMc) return;
  int d = t & (DMc - 1);
  int row = t >> 6;
  float sk = skip[0];
  const float* p = xn + (size_t)row * CIN;
  float* q = xm + (size_t)row * CIN;
  q[d]            = sk * p[d];
  q[DMc + d]      = sk * p[DMc + d];
  q[3 * DMc + d]  = (1.f + sk) * p[3 * DMc + d];
}

// ---------------------------------------------------------------------------
// Depthwise 3x3 (pad 1) on x3 slice of xn + BN + GELU -> dwout [NROWS x 64].
// ---------------------------------------------------------------------------
__global__ void k_dwconv(const float* __restrict__ xn, const float* __restrict__ dww,
                         const float* __restrict__ g, const float* __restrict__ bb,
                         const float* __restrict__ mm, const float* __restrict__ vv,
                         float* __restrict__ out) {
  int t = blockIdx.x * blockDim.x + threadIdx.x;   // NROWS * DM
  if (t >= NROWS * DMc) return;
  int c = t & (DMc - 1);
  int row = t >> 6;
  int l = row & (LL - 1);
  int b = row >> 12;
  int hy = l >> 6, wx = l & 63;
  float acc = 0.f;
#pragma unroll
  for (int dy = 0; dy < 3; ++dy) {
    int hh = hy + dy - 1;
    if (hh < 0 || hh >= HH) continue;
#pragma unroll
    for (int dx = 0; dx < 3; ++dx) {
      int wp = wx + dx - 1;
      if (wp < 0 || wp >= WWI) continue;
      acc += dww[c * 9 + dy * 3 + dx] *
             xn[((size_t)(b * LL + hh * WWI + wp)) * CIN + 2 * DMc + c];
    }
  }
  float bn = (acc - mm[c]) * rsqrtf(vv[c] + EPSF) * g[c] + bb[c];
  out[t] = geluf(bn);
}

// ---------------------------------------------------------------------------
// Pointwise epilogue: xm chunk2 = gelu(bn(pwout)) + skip * x3.
// ---------------------------------------------------------------------------
__global__ void k_pw_epi(const float* __restrict__ pw, const float* __restrict__ xn,
                         const float* __restrict__ g, const float* __restrict__ bb,
                         const float* __restrict__ mm, const float* __restrict__ vv,
                         const float* __restrict__ skip, float* __restrict__ xm) {
  int t = blockIdx.x * blockDim.x + threadIdx.x;   // NROWS * DM
  if (t >= NROWS * DMc) return;
  int c = t & (DMc - 1);
  int row = t >> 6;
  float bn = (pw[t] - mm[c]) * rsqrtf(vv[c] + EPSF) * g[c] + bb[c];
  xm[(size_t)row * CIN + 2 * DMc + c] =
      geluf(bn) + skip[0] * xn[(size_t)row * CIN + 2 * DMc + c];
}

// ---------------------------------------------------------------------------
extern "C" void kernel_launch(void* const* d_in, const int* in_sizes, int n_in,
                              void* d_out, int out_size, void* d_ws, size_t ws_size,
                              hipStream_t stream) {
  (void)in_sizes; (void)n_in; (void)out_size; (void)ws_size;
  const float* x          = (const float*)d_in[0];
  const float* ln_g       = (const float*)d_in[1];
  const float* ln_b       = (const float*)d_in[2];
  const float* in_proj_w  = (const float*)d_in[3];
  const float* conv1d_w   = (const float*)d_in[4];
  const float* conv1d_b   = (const float*)d_in[5];
  const float* x_proj_w   = (const float*)d_in[6];
  const float* dt_proj_w  = (const float*)d_in[7];
  const float* dt_proj_b  = (const float*)d_in[8];
  const float* A_log      = (const float*)d_in[9];
  const float* Dvec       = (const float*)d_in[10];
  const float* out_proj_w = (const float*)d_in[11];
  const float* dw_w       = (const float*)d_in[12];
  const float* bn_dw_g    = (const float*)d_in[13];
  const float* bn_dw_b    = (const float*)d_in[14];
  const float* bn_dw_m    = (const float*)d_in[15];
  const float* bn_dw_v    = (const float*)d_in[16];
  const float* pw_w       = (const float*)d_in[17];
  const float* bn_pw_g    = (const float*)d_in[18];
  const float* bn_pw_b    = (const float*)d_in[19];
  const float* bn_pw_m    = (const float*)d_in[20];
  const float* bn_pw_v    = (const float*)d_in[21];
  const float* proj_w     = (const float*)d_in[22];
  const float* proj_b     = (const float*)d_in[23];
  const float* skip       = (const float*)d_in[24];
  float* out = (float*)d_out;

  float* ws = (float*)d_ws;
  const size_t NLC = (size_t)NROWS * CIN;
  const size_t NSTATE = (size_t)BSZ * DIc * DSc * SSEG;   // 524288
  float* xn    = ws;                                  // [NROWS x 256]
  float* xm    = ws + NLC;                            // [NROWS x 256]
  float* xzb   = ws + 2 * NLC;                        // [NROWS x 256]
  float* xconv = ws + 3 * NLC;                        // [NROWS x 128]
  float* fused = xconv + (size_t)NROWS * DIc;         // [NROWS x 160]
  float* yg    = fused + (size_t)NROWS * NFUSE;       // [NROWS x 128]
  float* WinF  = yg + (size_t)NROWS * DIc;            // [256 x 64]
  float* WoF   = WinF + 2 * DIc * DMc;                // [64 x 128]
  float* Wcomb = WoF + DMc * DIc;                     // [160 x 128]
  float* bcomb = Wcomb + NFUSE * DIc;                 // [160]
  float* Pbuf  = bcomb + NFUSE;                       // [B*DI*DS*SSEG]
  float* qbuf  = Pbuf + NSTATE;
  float* H0buf = qbuf + NSTATE;
  float* dwout = xconv;                               // reuse after mamba
  float* pwout = yg;                                  // reuse after mamba
  float* xmln  = xzb;                                 // reuse after mamba

  // 1. LN over channels of x [B,C,L] -> xn [B,L,C]
  k_ln<<<NROWS / 256, 256, 0, stream>>>(x, ln_g, ln_b, xn, CIN * LL, LL, 1);
  // 2. flipped / fused weights
  k_prep<<<(2 * DIc * DMc + DMc * DIc + NFUSE * DIc + NFUSE + 255) / 256, 256, 0, stream>>>(
      in_proj_w, out_proj_w, x_proj_w, dt_proj_w, dt_proj_b, WinF, WoF, Wcomb, bcomb);
  // 3. skip-term init of xm chunks 0,1,3
  k_init_xm<<<(NROWS * DMc) / 256, 256, 0, stream>>>(xn, skip, xm);

  const int GM = NROWS / 32;                 // 1024 macro-rows (MB=2)
  // 4. four mamba instances: (x1,W), (x1,Wflip), (x2,W), (x2,Wflip)
  const float* Wins[4]  = { in_proj_w, WinF, in_proj_w, WinF };
  const float* Wouts[4] = { out_proj_w, WoF, out_proj_w, WoF };
  const int    coff[4]  = { 0, 0, DMc, DMc };
  for (int call = 0; call < 4; ++call) {
    const float* u = xn + coff[call];
    // in_proj: xz = u @ Win^T   [32768x64]x[64x256]; N=256 -> NB=4
    k_gemm_wmma<2, 4><<<GM * (2 * DIc / 64) / 8, 256, 0, stream>>>(
        u, Wins[call], nullptr, xzb, NROWS, 2 * DIc, DMc, CIN, DMc, 2 * DIc, 0);
    // causal depthwise conv1d + silu
    k_conv1d<<<(NROWS * DIc) / 256, 256, 0, stream>>>(xzb, conv1d_w, conv1d_b, xconv);
    // fused dt/x projection: [32768x128]x[128x160] + bias; N=160 -> NB=2
    k_gemm_wmma<2, 2><<<GM * (NFUSE / 32) / 8, 256, 0, stream>>>(
        xconv, Wcomb, bcomb, fused, NROWS, NFUSE, DIc, DIc, DIc, NFUSE, 1);
    // chunked parallel selective scan (3 passes) + D*x + silu(z) gate
    k_scan_p1<<<BSZ * (DIc / 2) * SSEG, 32, 0, stream>>>(xconv, fused, A_log, Pbuf, qbuf);
    k_scan_p2<<<BSZ * (DIc / 2), 32, 0, stream>>>(Pbuf, qbuf, H0buf);
    k_scan_p3<<<BSZ * (DIc / 2) * SSEG, 32, 0, stream>>>(xconv, fused, xzb, A_log, Dvec,
                                                         H0buf, yg);
    // out_proj, accumulate into xm chunk: [32768x128]x[128x64]; N=64 -> NB=4
    k_gemm_wmma<2, 4><<<GM * (DMc / 64) / 8, 256, 0, stream>>>(
        yg, Wouts[call], nullptr, xm + coff[call], NROWS, DMc, DIc, DIc, DIc, CIN, 2);
  }

  // 5. sepconv path on x3
  k_dwconv<<<(NROWS * DMc) / 256, 256, 0, stream>>>(
      xn, dw_w, bn_dw_g, bn_dw_b, bn_dw_m, bn_dw_v, dwout);
  k_gemm_wmma<2, 4><<<GM * (DMc / 64) / 8, 256, 0, stream>>>(
      dwout, pw_w, nullptr, pwout, NROWS, DMc, DMc, DMc, DMc, DMc, 0);
  k_pw_epi<<<(NROWS * DMc) / 256, 256, 0, stream>>>(
      pwout, xn, bn_pw_g, bn_pw_b, bn_pw_m, bn_pw_v, skip, xm);

  // 6. second LN (same gamma/beta) + final projection with transposed store
  k_ln<<<NROWS / 256, 256, 0, stream>>>(xm, ln_g, ln_b, xmln, LL * CIN, 1, CIN);
  k_gemm_wmma<2, 4><<<GM * (COUTC / 64) / 8, 256, 0, stream>>>(
      xmln, proj_w, proj_b, out, NROWS, COUTC, CIN, CIN, CIN, 0, 1 | 4);
}